// BayesianNN_19301583028478
// MI455X (gfx1250) — compile-verified
//
#include <hip/hip_runtime.h>
#include <cstdint>
#include <cstddef>

// ---------------------------------------------------------------------------
// Bayesian LSTM for MI455X (gfx1250, wave32).
//   B=256, T=2048, D=32, H=20, 4H=80  (hardcoded in the reference)
//
// Kernel 1: sample weights (mu + log1p(exp(rho))*eps) into workspace.
// Kernel 2: fused LSTM scan. 16 workgroups (one per 16-row batch tile),
//           5 waves each (one per 16-col gate tile of the 80-wide gates).
//
//           Software-pipelined recurrence:
//             acc_x(t+1) = bias + x_{t+1} @ w_ih   (8x WMMA, K=32)
//           is computed INSIDE the split barrier (s_barrier_signal ...
//           s_barrier_wait) so only the 5 h@w_hh WMMAs (K=20) remain on the
//           sequential critical path. Weight B-fragments live in VGPRs for
//           all 2048 steps; cell state in registers; gates/h round-trip
//           through LDS; output head (dot over H=20) fused.
// ---------------------------------------------------------------------------

typedef __attribute__((ext_vector_type(2))) float v2f;
typedef __attribute__((ext_vector_type(8))) float v8f;

#define B_SZ 256
#define T_SZ 2048
#define D_SZ 32
#define H_SZ 20
#define G_SZ 80          /* 4*H */

// workspace layout (floats)
#define WS_WIH  0        /* [32][80] */
#define WS_WHH  2560     /* [20][80] */
#define WS_BIAS 4160     /* [80]     */

__device__ __forceinline__ float softplusf(float x) {
  return log1pf(__expf(x));
}
__device__ __forceinline__ float sigmoidf_fast(float x) {
  return 1.0f / (1.0f + __expf(-x));
}
__device__ __forceinline__ float tanhf_fast(float x) {
  float e = __expf(2.0f * x);
  return (e - 1.0f) / (e + 1.0f);
}

// split workgroup barrier (CDNA5): signal, overlap independent work, wait
__device__ __forceinline__ void wg_barrier_signal() {
  asm volatile("s_wait_dscnt 0x0\n\ts_barrier_signal -1" ::: "memory");
}
__device__ __forceinline__ void wg_barrier_wait() {
  asm volatile("s_barrier_wait -1" ::: "memory");
}

// ---------------------------------------------------------------------------
// Kernel 1: sample the Bayesian weights into workspace (tiny, one block).
// ---------------------------------------------------------------------------
__global__ void bnn_sample_weights(
    const float* __restrict__ wih_mu, const float* __restrict__ wih_rho,
    const float* __restrict__ wih_eps,
    const float* __restrict__ whh_mu, const float* __restrict__ whh_rho,
    const float* __restrict__ whh_eps,
    const float* __restrict__ b_mu,   const float* __restrict__ b_rho,
    const float* __restrict__ b_eps,
    float* __restrict__ ws)
{
  const int tid = threadIdx.x;
  for (int i = tid; i < D_SZ * G_SZ; i += blockDim.x)
    ws[WS_WIH + i] = wih_mu[i] + softplusf(wih_rho[i]) * wih_eps[i];
  for (int i = tid; i < H_SZ * G_SZ; i += blockDim.x)
    ws[WS_WHH + i] = whh_mu[i] + softplusf(whh_rho[i]) * whh_eps[i];
  for (int i = tid; i < G_SZ; i += blockDim.x)
    ws[WS_BIAS + i] = b_mu[i] + softplusf(b_rho[i]) * b_eps[i];
}

// x-projection for one timestep: acc = bias + x_t @ w_ih  (8x WMMA, K=32)
__device__ __forceinline__ v8f xproj(const float* __restrict__ xt,
                                     const v2f* __restrict__ bih,
                                     float biasv, int koff) {
  v8f acc;
#pragma unroll
  for (int r = 0; r < 8; ++r) acc[r] = biasv;
#pragma unroll
  for (int kk = 0; kk < 8; ++kk) {
    v2f a;
    a.x = xt[kk * 4 + koff + 0];
    a.y = xt[kk * 4 + koff + 1];
    acc = __builtin_amdgcn_wmma_f32_16x16x4_f32(
              false, a, false, bih[kk], (short)0, acc, false, false);
  }
  return acc;
}

// ---------------------------------------------------------------------------
// Kernel 2: fused LSTM scan + output head.
//
// WMMA f32 16x16x4 fragment layout assumptions (per CDNA5 ISA 7.12.2):
//   A (16x4):  lane L: M = L%16, half = L/16; vgpr v holds K = 2*half + v
//   B (4x16):  lane L: N = L%16, half = L/16; vgpr v holds K = 2*half + v
//   C/D (16x16): lane L: N = L%16, half = L/16; vgpr r holds M = r + 8*half
// ---------------------------------------------------------------------------
__launch_bounds__(160, 1)
__global__ void bnn_lstm_scan(const float* __restrict__ x,      // [B][T][D]
                              const float* __restrict__ ws,
                              const float* __restrict__ w_lin,  // [H]
                              const float* __restrict__ b_lin,  // [1]
                              float* __restrict__ out)          // [B][T]
{
  __shared__ float gates[16 * 84];     // 16 rows x 80 gates, stride 84 (bank pad)
  __shared__ float hbuf[16 * H_SZ];    // current hidden state, row-major
  __shared__ float wl[H_SZ];
  __shared__ float blin_s;

  const int tid  = threadIdx.x;
  const int wave = tid >> 5;           // 0..4  -> gate N-tile
  const int lane = tid & 31;
  const int n16  = lane & 15;
  const int half = lane >> 4;
  const int b0   = blockIdx.x * 16;    // batch tile base row

  const float* wih = ws + WS_WIH;      // [32][80]
  const float* whh = ws + WS_WHH;      // [20][80]
  const int col  = wave * 16 + n16;    // gate column owned by this lane
  const int koff = 2 * half;

  // ---- persistent weight B-fragments (live in VGPRs for the whole scan) ----
  v2f bih[8];
#pragma unroll
  for (int kk = 0; kk < 8; ++kk) {
    const int k = kk * 4 + koff;
    bih[kk].x = wih[(k + 0) * G_SZ + col];
    bih[kk].y = wih[(k + 1) * G_SZ + col];
  }
  v2f bhh[5];
#pragma unroll
  for (int kk = 0; kk < 5; ++kk) {
    const int k = kk * 4 + koff;
    bhh[kk].x = whh[(k + 0) * G_SZ + col];
    bhh[kk].y = whh[(k + 1) * G_SZ + col];
  }
  const float biasv = ws[WS_BIAS + col];

  // ---- stage head weights, zero hidden state ----
  if (tid < H_SZ) wl[tid] = w_lin[tid];
  if (tid == 0)  blin_s = b_lin[0];
  for (int i = tid; i < 16 * H_SZ; i += 160) hbuf[i] = 0.0f;

  // per-thread cell state: 320 (row,unit) pairs over 160 threads -> 2 each
  float c0 = 0.0f, c1 = 0.0f;
  const int p0 = tid,        r0 = p0 / H_SZ, u0 = p0 % H_SZ;
  const int p1 = tid + 160,  r1 = p1 / H_SZ, u1 = p1 % H_SZ;

  // per-lane x A-fragment row base (row M = b0 + n16); one x row = 128B line
  const size_t xrow = (size_t)(b0 + n16) * T_SZ * D_SZ;

  __syncthreads();

  // ---- software-pipeline prologue: x-projection for t = 0 ----
  v8f acc_x = xproj(x + xrow, bih, biasv, koff);

  for (int t = 0; t < T_SZ; ++t) {
    // --- sequential critical path: h @ w_hh (5x WMMA, K=20) ---
    v8f acc = acc_x;
#pragma unroll
    for (int kk = 0; kk < 5; ++kk) {
      v2f a;
      a.x = hbuf[n16 * H_SZ + kk * 4 + koff + 0];
      a.y = hbuf[n16 * H_SZ + kk * 4 + koff + 1];
      acc = __builtin_amdgcn_wmma_f32_16x16x4_f32(
                false, a, false, bhh[kk], (short)0, acc, false, false);
    }

    // scatter gates tile to LDS: row M = r + 8*half
#pragma unroll
    for (int r = 0; r < 8; ++r)
      gates[(r + 8 * half) * 84 + col] = acc[r];

    // --- split barrier 1: overlap next-step x-projection with the barrier ---
    wg_barrier_signal();
    if (t + 1 < T_SZ)
      acc_x = xproj(x + xrow + (size_t)(t + 1) * D_SZ, bih, biasv, koff);
    // prefetch the x rows two steps ahead (one 128B cacheline per lane)
    if (wave == 0 && lane < 16 && (t + 2) < T_SZ)
      __builtin_prefetch(&x[(size_t)(b0 + lane) * T_SZ * D_SZ +
                            (size_t)(t + 2) * D_SZ], 0, 3);
    wg_barrier_wait();

    // --- LSTM cell elementwise: gates layout [i | f | g | o], each H wide ---
    {
      const float* g = &gates[r0 * 84];
      float i_ = sigmoidf_fast(g[u0]);
      float f_ = sigmoidf_fast(g[H_SZ + u0]);
      float g_ = tanhf_fast(g[2 * H_SZ + u0]);
      float o_ = sigmoidf_fast(g[3 * H_SZ + u0]);
      c0 = f_ * c0 + i_ * g_;
      hbuf[r0 * H_SZ + u0] = o_ * tanhf_fast(c0);
    }
    {
      const float* g = &gates[r1 * 84];
      float i_ = sigmoidf_fast(g[u1]);
      float f_ = sigmoidf_fast(g[H_SZ + u1]);
      float g_ = tanhf_fast(g[2 * H_SZ + u1]);
      float o_ = sigmoidf_fast(g[3 * H_SZ + u1]);
      c1 = f_ * c1 + i_ * g_;
      hbuf[r1 * H_SZ + u1] = o_ * tanhf_fast(c1);
    }

    // --- split barrier 2: h ready for all waves ---
    wg_barrier_signal();
    wg_barrier_wait();

    // fused output head: out[b][t] = h . w_lin + b_lin  (16 rows)
    if (tid < 16) {
      float s = blin_s;
#pragma unroll
      for (int u = 0; u < H_SZ; ++u) s += hbuf[tid * H_SZ + u] * wl[u];
      out[(size_t)(b0 + tid) * T_SZ + t] = s;
    }
  }
}

// ---------------------------------------------------------------------------
// Host launcher
// ---------------------------------------------------------------------------
extern "C" void kernel_launch(void* const* d_in, const int* in_sizes, int n_in,
                              void* d_out, int out_size, void* d_ws, size_t ws_size,
                              hipStream_t stream) {
  (void)in_sizes; (void)n_in; (void)out_size; (void)ws_size;

  const float* x        = (const float*)d_in[0];
  const float* wih_mu   = (const float*)d_in[1];
  const float* wih_rho  = (const float*)d_in[2];
  const float* wih_eps  = (const float*)d_in[3];
  const float* whh_mu   = (const float*)d_in[4];
  const float* whh_rho  = (const float*)d_in[5];
  const float* whh_eps  = (const float*)d_in[6];
  const float* b_mu     = (const float*)d_in[7];
  const float* b_rho    = (const float*)d_in[8];
  const float* b_eps    = (const float*)d_in[9];
  const float* w_lin    = (const float*)d_in[10];
  const float* b_lin    = (const float*)d_in[11];
  float*       out      = (float*)d_out;
  float*       ws       = (float*)d_ws;

  bnn_sample_weights<<<1, 256, 0, stream>>>(
      wih_mu, wih_rho, wih_eps, whh_mu, whh_rho, whh_eps,
      b_mu, b_rho, b_eps, ws);

  bnn_lstm_scan<<<B_SZ / 16, 160, 0, stream>>>(x, ws, w_lin, b_lin, out);
}